// CellGraphSignatureGNN_11072425689891
// MI455X (gfx1250) — compile-verified
//
#include <hip/hip_runtime.h>

typedef __attribute__((ext_vector_type(2))) float v2f;
typedef __attribute__((ext_vector_type(8))) float v8f;

#define WIDTH 128
#define NGRAPH 64

// ---------------- degree / norm precompute ----------------

__global__ __launch_bounds__(256) void k_init_deg(float* deg, int n) {
    int i = blockIdx.x * 256 + threadIdx.x;
    if (i < n) deg[i] = 2.0f;   // self-loop fill_value = 2.0 (improved GCN)
}

__global__ __launch_bounds__(256) void k_deg_edges(const int* __restrict__ col,
                                                   const float* __restrict__ w,
                                                   float* deg, int e) {
    int i = blockIdx.x * 256 + threadIdx.x;
    if (i < e) atomicAdd(&deg[col[i]], w[i]);
}

__global__ __launch_bounds__(256) void k_dinv(float* deg, int n) {
    int i = blockIdx.x * 256 + threadIdx.x;
    if (i < n) {
        float d = deg[i];
        deg[i] = (d > 0.0f) ? rsqrtf(d) : 0.0f;
    }
}

__global__ __launch_bounds__(256) void k_norm(const int* __restrict__ row,
                                              const int* __restrict__ col,
                                              const float* __restrict__ w,
                                              const float* __restrict__ dinv,
                                              float* __restrict__ norm, int e) {
    int i = blockIdx.x * 256 + threadIdx.x;
    if (i < e) norm[i] = dinv[row[i]] * w[i] * dinv[col[i]];
}

__global__ __launch_bounds__(256) void k_selfnorm(const float* __restrict__ dinv,
                                                  float* __restrict__ sn, int n) {
    int i = blockIdx.x * 256 + threadIdx.x;
    if (i < n) { float d = dinv[i]; sn[i] = 2.0f * d * d; }
}

// ---------------- dense GEMM: H = X @ W  (fp32 WMMA 16x16x4) ----------------
// Block = 256 threads = 8 waves; computes a 128x128 output block.
// Wave w owns rows [w*16, w*16+16), iterates 8 column tiles of 16.
// A 16x4 frag: lane(0..15)->M, lane[4] selects K-pair; one v2f (ds_load_b64).
// B 4x16 frag: lane(0..15)->N; W staged K-pair-interleaved in LDS so each
// fragment is a single aligned 8-byte LDS load straight into the WMMA pair.

__global__ __launch_bounds__(256) void k_gemm128(const float* __restrict__ X,
                                                 const float* __restrict__ W,
                                                 float* __restrict__ H, int n) {
    __shared__ float lA[128 * WIDTH];     // 64 KB  A-stripe, row-major (zero-padded)
    __shared__ float lB[WIDTH * WIDTH];   // 64 KB  W, K-pair interleaved:
                                          // lB[k2*256 + n*2 + p] = W[2*k2+p][n]

    const int tid = threadIdx.x;
    const int rowBase = blockIdx.x * 128;

    // stage W interleaved: read float4 of a row, scatter 4 scalars
    for (int i = tid; i < (WIDTH * WIDTH) / 4; i += 256) {
        float4 v = ((const float4*)W)[i];
        int r = i >> 5;                 // W row (= K index), 32 float4 per row
        int c = (i & 31) << 2;          // column base
        float* dst = &lB[(r >> 1) * 256 + (r & 1)];
        dst[(c + 0) * 2] = v.x;
        dst[(c + 1) * 2] = v.y;
        dst[(c + 2) * 2] = v.z;
        dst[(c + 3) * 2] = v.w;
    }

    // stage A block, zero-pad out-of-range rows
    for (int i = tid; i < (128 * WIDTH) / 4; i += 256) {
        int r = i >> 5;                 // 32 float4 per 128-float row
        int gr = rowBase + r;
        float4 v = make_float4(0.0f, 0.0f, 0.0f, 0.0f);
        if (gr < n) v = ((const float4*)X)[gr * 32 + (i & 31)];
        ((float4*)lA)[i] = v;
    }
    __syncthreads();

    const int wave = tid >> 5;
    const int lane = tid & 31;
    const int m  = lane & 15;             // M for A-frag, N for B-frag
    const int kb = (lane >> 4) << 1;      // K-pair base: 0 or 2

    const float* Arow = &lA[(wave * 16 + m) * WIDTH + kb];
    const v2f*   Bp   = (const v2f*)lB;   // Bp[k2*128 + n] = {W[2k2][n], W[2k2+1][n]}

    v8f acc[8] = {};

    for (int k = 0; k < WIDTH; k += 4) {
        v2f a = *(const v2f*)(Arow + k);              // {A[m][k+kb], A[m][k+kb+1]}
        const v2f* Bk = &Bp[((k + kb) >> 1) * 128 + m];
#pragma unroll
        for (int t = 0; t < 8; ++t) {
            v2f b = Bk[t * 16];                       // {W[k+kb][n], W[k+kb+1][n]}
            acc[t] = __builtin_amdgcn_wmma_f32_16x16x4_f32(
                false, a, false, b, (short)0, acc[t], false, false);
        }
    }

    // C/D layout: VGPR v -> row (v) for lanes 0-15, row (v+8) for lanes 16-31
    const int rowTop = rowBase + wave * 16 + ((lane < 16) ? 0 : 8);
    if (rowBase + 128 <= n) {
        // fast path: whole block in range
#pragma unroll
        for (int t = 0; t < 8; ++t) {
#pragma unroll
            for (int v = 0; v < 8; ++v)
                H[(rowTop + v) * WIDTH + t * 16 + m] = acc[t][v];
        }
    } else {
#pragma unroll
        for (int t = 0; t < 8; ++t) {
#pragma unroll
            for (int v = 0; v < 8; ++v) {
                int gr = rowTop + v;
                if (gr < n) H[gr * WIDTH + t * 16 + m] = acc[t][v];
            }
        }
    }
}

// ---------------- bias init + self-loop (single writer per node) ----------------

__global__ __launch_bounds__(256) void k_node_init(const float* __restrict__ H,
                                                   const float* __restrict__ sn,
                                                   const float* __restrict__ bias,
                                                   float* __restrict__ out, int n) {
    int i = blockIdx.x * 8 + (threadIdx.x >> 5);
    if (i >= n) return;
    int lane = threadIdx.x & 31;
    float s = sn[i];
    float4 h = ((const float4*)(H + (size_t)i * WIDTH))[lane];
    float4 b = ((const float4*)bias)[lane];
    float4 o;
    o.x = b.x + s * h.x;
    o.y = b.y + s * h.y;
    o.z = b.z + s * h.z;
    o.w = b.w + s * h.w;
    ((float4*)(out + (size_t)i * WIDTH))[lane] = o;
}

// ---------------- edge gather-scale-scatter (one wave per edge) ----------------

__global__ __launch_bounds__(256) void k_scatter(const float* __restrict__ H,
                                                 const float* __restrict__ norm,
                                                 const int* __restrict__ row,
                                                 const int* __restrict__ col,
                                                 float* __restrict__ out, int e) {
    int eid = blockIdx.x * 8 + (threadIdx.x >> 5);
    if (eid >= e) return;
    int lane = threadIdx.x & 31;
    int r = row[eid];
    int c = col[eid];
    float nv = norm[eid];
    float4 h = ((const float4*)(H + (size_t)r * WIDTH))[lane];
    float* o = out + (size_t)c * WIDTH + lane * 4;
    atomicAdd(o + 0, nv * h.x);
    atomicAdd(o + 1, nv * h.y);
    atomicAdd(o + 2, nv * h.z);
    atomicAdd(o + 3, nv * h.w);
}

// ---------------- global mean pool ----------------

__global__ __launch_bounds__(256) void k_pool_zero(float* sums, float* counts) {
    int i = blockIdx.x * 256 + threadIdx.x;
    if (i < NGRAPH * WIDTH) sums[i] = 0.0f;
    if (i < NGRAPH) counts[i] = 0.0f;
}

__global__ __launch_bounds__(256) void k_pool_sum(const float* __restrict__ X,
                                                  const int* __restrict__ batch,
                                                  float* sums, float* counts, int n) {
    int i = blockIdx.x * 8 + (threadIdx.x >> 5);
    if (i >= n) return;
    int lane = threadIdx.x & 31;
    int g = batch[i];
    float4 v = ((const float4*)(X + (size_t)i * WIDTH))[lane];
    float* s = sums + g * WIDTH + lane * 4;
    atomicAdd(s + 0, v.x);
    atomicAdd(s + 1, v.y);
    atomicAdd(s + 2, v.z);
    atomicAdd(s + 3, v.w);
    if (lane == 0) atomicAdd(&counts[g], 1.0f);
}

__global__ __launch_bounds__(256) void k_pool_div(const float* __restrict__ sums,
                                                  const float* __restrict__ counts,
                                                  float* __restrict__ out) {
    int i = blockIdx.x * 256 + threadIdx.x;
    if (i < NGRAPH * WIDTH) out[i] = sums[i] / fmaxf(counts[i >> 7], 1.0f);
}

// ---------------- host orchestration ----------------

extern "C" void kernel_launch(void* const* d_in, const int* in_sizes, int n_in,
                              void* d_out, int out_size, void* d_ws, size_t ws_size,
                              hipStream_t stream) {
    const float* x    = (const float*)d_in[0];
    const int*   ei   = (const int*)d_in[1];
    const float* ew   = (const float*)d_in[2];
    const int*   batch= (const int*)d_in[3];
    const float* Ws   = (const float*)d_in[4];
    const float* bs   = (const float*)d_in[5];

    const int N = in_sizes[0] / WIDTH;
    const int E = in_sizes[1] / 2;
    const int* row = ei;
    const int* col = ei + E;

    char* ws = (char*)d_ws;
    size_t off = 0;
    auto alloc = [&](size_t bytes) -> void* {
        void* p = ws + off;
        off += (bytes + 255) & ~(size_t)255;
        return p;
    };

    float* H      = (float*)alloc((size_t)N * WIDTH * 4);
    float* Bx     = (float*)alloc((size_t)N * WIDTH * 4);
    float* Cx     = (float*)alloc((size_t)N * WIDTH * 4);
    float* dinv   = (float*)alloc((size_t)N * 4);      // deg -> dinv in place
    float* norm   = (float*)alloc((size_t)E * 4);
    float* sn     = (float*)alloc((size_t)N * 4);
    float* sums   = (float*)alloc((size_t)NGRAPH * WIDTH * 4);
    float* counts = (float*)alloc((size_t)NGRAPH * 4);

    const int nb_n  = (N + 255) / 256;
    const int nb_e  = (E + 255) / 256;
    const int nb_g  = (N + 127) / 128;   // gemm blocks
    const int nb_nw = (N + 7) / 8;       // one wave per node
    const int nb_ew = (E + 7) / 8;       // one wave per edge

    // degrees & normalization (done once)
    k_init_deg <<<nb_n, 256, 0, stream>>>(dinv, N);
    k_deg_edges<<<nb_e, 256, 0, stream>>>(col, ew, dinv, E);
    k_dinv     <<<nb_n, 256, 0, stream>>>(dinv, N);
    k_norm     <<<nb_e, 256, 0, stream>>>(row, col, ew, dinv, norm, E);
    k_selfnorm <<<nb_n, 256, 0, stream>>>(dinv, sn, N);

    // layer 0: x -> Bx
    k_gemm128  <<<nb_g,  256, 0, stream>>>(x, Ws + 0 * WIDTH * WIDTH, H, N);
    k_node_init<<<nb_nw, 256, 0, stream>>>(H, sn, bs + 0 * WIDTH, Bx, N);
    k_scatter  <<<nb_ew, 256, 0, stream>>>(H, norm, row, col, Bx, E);

    // layer 1: Bx -> Cx
    k_gemm128  <<<nb_g,  256, 0, stream>>>(Bx, Ws + 1 * WIDTH * WIDTH, H, N);
    k_node_init<<<nb_nw, 256, 0, stream>>>(H, sn, bs + 1 * WIDTH, Cx, N);
    k_scatter  <<<nb_ew, 256, 0, stream>>>(H, norm, row, col, Cx, E);

    // layer 2: Cx -> Bx
    k_gemm128  <<<nb_g,  256, 0, stream>>>(Cx, Ws + 2 * WIDTH * WIDTH, H, N);
    k_node_init<<<nb_nw, 256, 0, stream>>>(H, sn, bs + 2 * WIDTH, Bx, N);
    k_scatter  <<<nb_ew, 256, 0, stream>>>(H, norm, row, col, Bx, E);

    // global mean pool
    k_pool_zero<<<(NGRAPH * WIDTH + 255) / 256, 256, 0, stream>>>(sums, counts);
    k_pool_sum <<<nb_nw, 256, 0, stream>>>(Bx, batch, sums, counts, N);
    k_pool_div <<<(NGRAPH * WIDTH + 255) / 256, 256, 0, stream>>>(sums, counts, (float*)d_out);
}